// RNN_LSTM_1589137900338
// MI455X (gfx1250) — compile-verified
//
#include <hip/hip_runtime.h>
#include <hip/hip_bf16.h>

// Problem dims (fixed by the reference)
#define T_STEPS 4096
#define DIM_I   2048
#define DIM_H   2048
#define DIM_4H  8192

typedef __attribute__((ext_vector_type(2))) float v2f;
typedef __attribute__((ext_vector_type(8))) float v8f;
typedef __attribute__((ext_vector_type(4))) int   v4i;

#define AS_GLOBAL __attribute__((address_space(1)))
#define AS_LOCAL  __attribute__((address_space(3)))

// ---------------------------------------------------------------------------
// Fragment loader for V_WMMA_F32_16X16X4_F32.
// A-matrix 16x4 layout (ISA 7.12.2): lanes 0-15 -> M=lane, VGPR0=K0,VGPR1=K1;
// lanes 16-31 -> M=lane-16, VGPR0=K2, VGPR1=K3.  B (4x16) is the symmetric
// transpose layout, so loading B from a row-major [N,K] matrix uses the exact
// same addressing with the N-tile base as "row0".
// ---------------------------------------------------------------------------
__device__ __forceinline__ v2f load_frag_f32(const float* __restrict__ p,
                                             int row0, int ldk, int k, int lane) {
  const int l  = lane & 15;
  const int hi = lane >> 4;               // 0 or 1 -> selects K pair
  const float* q = p + (size_t)(row0 + l) * ldk + k + 2 * hi;
  v2f r;
  r.x = q[0];
  r.y = q[1];
  return r;
}

__device__ __forceinline__ float sigmoid_f(float x) {
  return 1.0f / (1.0f + __expf(-x));
}

// ---------------------------------------------------------------------------
// C[M,N] = A[M,K] @ B[N,K]^T + bias1[N] (+ bias2[N]), optional sigmoid.
// Wave tile: 32(M) x 64(N); block: 8 waves = 64(M) x 256(N).
// FP32 WMMA (16x16x4) keeps full fp32 fidelity vs the reference; both GEMM
// operands stay resident in the 192 MB L2.
// ---------------------------------------------------------------------------
template <bool SIGMOID>
__global__ __launch_bounds__(256)
void wmma_gemm_nt(const float* __restrict__ A, const float* __restrict__ B,
                  const float* __restrict__ bias1, const float* __restrict__ bias2,
                  float* __restrict__ C, int M, int N, int K) {
  const int lane = threadIdx.x & 31;
  const int wave = threadIdx.x >> 5;      // 0..7
  const int m0 = blockIdx.y * 64  + (wave >> 2) * 32;   // 2 waves stacked in M
  const int n0 = blockIdx.x * 256 + (wave & 3) * 64;    // 4 waves across N

  v8f acc[2][4] = {};

  for (int k = 0; k < K; k += 4) {
    const v2f a0 = load_frag_f32(A, m0,      K, k, lane);
    const v2f a1 = load_frag_f32(A, m0 + 16, K, k, lane);
    const v2f b0 = load_frag_f32(B, n0,      K, k, lane);
    const v2f b1 = load_frag_f32(B, n0 + 16, K, k, lane);
    const v2f b2 = load_frag_f32(B, n0 + 32, K, k, lane);
    const v2f b3 = load_frag_f32(B, n0 + 48, K, k, lane);

    acc[0][0] = __builtin_amdgcn_wmma_f32_16x16x4_f32(false, a0, false, b0, (short)0, acc[0][0], false, false);
    acc[0][1] = __builtin_amdgcn_wmma_f32_16x16x4_f32(false, a0, false, b1, (short)0, acc[0][1], false, false);
    acc[0][2] = __builtin_amdgcn_wmma_f32_16x16x4_f32(false, a0, false, b2, (short)0, acc[0][2], false, false);
    acc[0][3] = __builtin_amdgcn_wmma_f32_16x16x4_f32(false, a0, false, b3, (short)0, acc[0][3], false, false);
    acc[1][0] = __builtin_amdgcn_wmma_f32_16x16x4_f32(false, a1, false, b0, (short)0, acc[1][0], false, false);
    acc[1][1] = __builtin_amdgcn_wmma_f32_16x16x4_f32(false, a1, false, b1, (short)0, acc[1][1], false, false);
    acc[1][2] = __builtin_amdgcn_wmma_f32_16x16x4_f32(false, a1, false, b2, (short)0, acc[1][2], false, false);
    acc[1][3] = __builtin_amdgcn_wmma_f32_16x16x4_f32(false, a1, false, b3, (short)0, acc[1][3], false, false);
  }

  // C/D 16x16 layout: lanes 0-15 -> N=lane, VGPR v -> M=v; lanes 16-31 -> N=lane-16, M=v+8
  const int l  = lane & 15;
  const int hi = lane >> 4;
#pragma unroll
  for (int mi = 0; mi < 2; ++mi) {
#pragma unroll
    for (int ni = 0; ni < 4; ++ni) {
      const int col = n0 + ni * 16 + l;
      float b = bias1[col];
      if (bias2) b += bias2[col];
#pragma unroll
      for (int v = 0; v < 8; ++v) {
        const int row = m0 + mi * 16 + v + 8 * hi;
        float val = acc[mi][ni][v] + b;
        if (SIGMOID) val = sigmoid_f(val);
        C[(size_t)row * N + col] = val;
      }
    }
  }
}

// ---------------------------------------------------------------------------
// Grid-wide barrier (32 blocks, safely co-resident) for the persistent scan.
// ---------------------------------------------------------------------------
__device__ __forceinline__ void grid_barrier(unsigned* counter, unsigned* gen,
                                             unsigned nblk) {
  __syncthreads();
  if (threadIdx.x == 0) {
    __threadfence();  // make this block's h writes visible at device scope
    const unsigned g = __hip_atomic_load(gen, __ATOMIC_RELAXED, __HIP_MEMORY_SCOPE_AGENT);
    const unsigned arrived =
        __hip_atomic_fetch_add(counter, 1u, __ATOMIC_ACQ_REL, __HIP_MEMORY_SCOPE_AGENT);
    if (arrived == nblk - 1u) {
      __hip_atomic_store(counter, 0u, __ATOMIC_RELAXED, __HIP_MEMORY_SCOPE_AGENT);
      __hip_atomic_fetch_add(gen, 1u, __ATOMIC_RELEASE, __HIP_MEMORY_SCOPE_AGENT);
    } else {
      while (__hip_atomic_load(gen, __ATOMIC_ACQUIRE, __HIP_MEMORY_SCOPE_AGENT) == g) {
        __builtin_amdgcn_s_sleep(1);
      }
    }
  }
  __syncthreads();
}

// ---------------------------------------------------------------------------
// Stage h[0..2047] (global) into LDS. Prefer the gfx1250 ASYNCcnt-tracked
// global->LDS DMA path (no VGPR round-trip); fall back to a plain copy if
// this toolchain doesn't declare the async builtins.
// Builtin prototype (from hipcc diagnostic): (global v4i*, lds v4i*, Ii, Ii).
// ---------------------------------------------------------------------------
__device__ __forceinline__ void stage_h_to_lds(const float* __restrict__ h,
                                               float* hl, int tid) {
#if __has_builtin(__builtin_amdgcn_global_load_async_to_lds_b128)
  for (int i = tid * 4; i < DIM_H; i += 256 * 4) {
    __builtin_amdgcn_global_load_async_to_lds_b128(
        (AS_GLOBAL v4i*)(h + i), (AS_LOCAL v4i*)(hl + i), 0, 0);
  }
#if __has_builtin(__builtin_amdgcn_s_wait_asynccnt)
  __builtin_amdgcn_s_wait_asynccnt(0);
#else
  asm volatile("s_wait_asynccnt 0x0" ::: "memory");
#endif
#else
  for (int i = tid; i < DIM_H; i += 256) hl[i] = h[i];
#endif
  __syncthreads();
}

// ---------------------------------------------------------------------------
// Persistent LSTM scan: 32 blocks x 256 threads = 8192 lanes.
// Lane t: unit j = t>>2, gate = t&3 (i,f,g,o), row = gate*2048 + j.
// W_hh (64 MB fp32) stays resident in the 192 MB L2 across all 4096 steps.
// Gates of a unit sit in 4 adjacent lanes -> wave-local __shfl gather,
// so only ONE grid barrier per timestep. Cell state c lives in registers.
// ---------------------------------------------------------------------------
__global__ __launch_bounds__(256)
void lstm_scan(const float* __restrict__ Whh, const float* __restrict__ gx,
               float* __restrict__ h, float* __restrict__ hs,
               unsigned* counter, unsigned* gen, unsigned nblk) {
  const int tid  = threadIdx.x;
  const int t    = blockIdx.x * 256 + tid;   // 0..8191
  const int j    = t >> 2;                   // hidden unit
  const int gate = t & 3;                    // 0=i 1=f 2=g 3=o
  const int row  = gate * DIM_H + j;
  const float* __restrict__ wrow = Whh + (size_t)row * DIM_H;
  const int lane = tid & 31;
  const int base = lane & ~3;

  __shared__ __align__(16) float hl[DIM_H];

  float c = 0.0f;

  for (int step = 0; step < T_STEPS; ++step) {
    // stage h (zeros at step 0, from init kernel) into LDS via async DMA
    stage_h_to_lds(h, hl, tid);

    // warm L2 for the next step's gx element (only HBM-side read in the loop)
    if (step + 1 < T_STEPS)
      __builtin_prefetch(gx + (size_t)(step + 1) * DIM_4H + row, 0, 0);

    float acc = 0.0f;
#pragma unroll 4
    for (int k = 0; k < DIM_H; k += 4) {
      const float4 w  = *(const float4*)(wrow + k);
      const float4 hv = *(const float4*)(&hl[k]);
      acc += w.x * hv.x + w.y * hv.y + w.z * hv.z + w.w * hv.w;
    }
    const float g = acc + gx[(size_t)step * DIM_4H + row];

    // gather all 4 gates of this unit from adjacent lanes
    const float gi = __shfl(g, base + 0, 32);
    const float gf = __shfl(g, base + 1, 32);
    const float gg = __shfl(g, base + 2, 32);
    const float go = __shfl(g, base + 3, 32);

    const float c_new = sigmoid_f(gf) * c + sigmoid_f(gi) * tanhf(gg);
    const float h_new = sigmoid_f(go) * tanhf(c_new);
    c = c_new;

    if (gate == 0) {
      h[j] = h_new;
      hs[(size_t)step * DIM_H + j] = h_new;
    }
    grid_barrier(counter, gen, nblk);
  }
}

// Zero h + barrier state (ws is poisoned by the harness).
__global__ void scan_init(float* h, unsigned* counter, unsigned* gen) {
  const int i = threadIdx.x + blockIdx.x * blockDim.x;
  if (i < DIM_H) h[i] = 0.0f;
  if (i == 0) { *counter = 0u; *gen = 0u; }
}

extern "C" void kernel_launch(void* const* d_in, const int* in_sizes, int n_in,
                              void* d_out, int out_size, void* d_ws, size_t ws_size,
                              hipStream_t stream) {
  const float* x    = (const float*)d_in[0];  // [1,T,I]
  const float* W_ih = (const float*)d_in[1];  // [4H,I]
  const float* W_hh = (const float*)d_in[2];  // [4H,H]
  const float* b_ih = (const float*)d_in[3];  // [4H]
  const float* b_hh = (const float*)d_in[4];  // [4H]
  const float* Wo   = (const float*)d_in[5];  // [I,H]
  const float* bo   = (const float*)d_in[6];  // [I]
  float* out = (float*)d_out;                 // [1,T,I]

  // Workspace layout
  unsigned char* ws = (unsigned char*)d_ws;
  unsigned* counter = (unsigned*)ws;                       // 4 B
  unsigned* gen     = (unsigned*)(ws + 64);                // 4 B
  float* h  = (float*)(ws + 256);                          // 8 KB (16B aligned)
  float* gx = (float*)(ws + 256 + DIM_H * sizeof(float));  // T*4H fp32 = 128 MB
  float* hs = gx + (size_t)T_STEPS * DIM_4H;               // T*H  fp32 =  32 MB

  // 0) init h + barrier state
  scan_init<<<(DIM_H + 255) / 256, 256, 0, stream>>>(h, counter, gen);

  // 1) gx[T,4H] = x @ W_ih^T + (b_ih + b_hh)     (fp32 WMMA)
  {
    dim3 grid(DIM_4H / 256, T_STEPS / 64);
    wmma_gemm_nt<false><<<grid, 256, 0, stream>>>(x, W_ih, b_ih, b_hh, gx,
                                                  T_STEPS, DIM_4H, DIM_I);
  }

  // 2) sequential LSTM scan (persistent, one grid barrier per step)
  lstm_scan<<<32, 256, 0, stream>>>(W_hh, gx, h, hs, counter, gen, 32u);

  // 3) out = sigmoid(hs @ Wo^T + bo)             (fp32 WMMA + fused epilogue)
  {
    dim3 grid(DIM_I / 256, T_STEPS / 64);
    wmma_gemm_nt<true><<<grid, 256, 0, stream>>>(hs, Wo, bo, nullptr, out,
                                                 T_STEPS, DIM_I, DIM_H);
  }
}